// NLFMax_15796889715110
// MI455X (gfx1250) — compile-verified
//
#include <hip/hip_runtime.h>

// ---------------------------------------------------------------------------
// NLF max: four directional linear scans + elementwise max.
// [B,C,H,W] = [4,16,320,640] fp32.  Memory-bound (~314MB min traffic).
// ---------------------------------------------------------------------------

#define B_ 4
#define C_ 16
#define H_ 320
#define W_ 640
#define NPLANES (B_ * C_)      // 64
#define PLANE   (H_ * W_)      // 204800
#define NROWS   (NPLANES * H_) // 20480

// ---------------- CDNA5 async global->LDS staging (probe-confirmed) --------
#if defined(__gfx1250__) && \
    __has_builtin(__builtin_amdgcn_global_load_async_to_lds_b128) && \
    __has_builtin(__builtin_amdgcn_s_wait_asynccnt)
#define NLF_ASYNC_LDS 1
#else
#define NLF_ASYNC_LDS 0
#endif

#if NLF_ASYNC_LDS
typedef int v4i __attribute__((vector_size(16)));
typedef __attribute__((address_space(1))) v4i gv4i;  // global int4
typedef __attribute__((address_space(3))) v4i lv4i;  // LDS int4
#endif

__device__ __forceinline__ void copy16_to_lds(void* lds, const void* glob) {
#if NLF_ASYNC_LDS
  // GLOBAL_LOAD_ASYNC_TO_LDS_B128 (tracked by ASYNCcnt)
  __builtin_amdgcn_global_load_async_to_lds_b128(
      (gv4i*)glob, (lv4i*)lds, /*imm offset*/0, /*cpol*/0);
#else
  *reinterpret_cast<float4*>(lds) = *reinterpret_cast<const float4*>(glob);
#endif
}

__device__ __forceinline__ void staging_fence() {
#if NLF_ASYNC_LDS
  __builtin_amdgcn_s_wait_asynccnt(0);   // s_wait_asynccnt 0
#endif
  __syncthreads();
}

// ---------------------------------------------------------------------------
// Vertical kernel: down (g0) and up (g1) scans over H, out = max(down, up).
// Block = one plane x 32 columns.  8 waves, each owns a 40-row segment.
// Blocked affine scan: segment composes to f(p) = A*p + B,
// compose f2∘f1 => (A2*A1, A2*B1 + B2).
// ---------------------------------------------------------------------------
#define VT_W   32
#define VT_SEG (H_ / 8)   // 40

__global__ __launch_bounds__(256) void nlf_vertical(
    const float* __restrict__ x, const float* __restrict__ g0,
    const float* __restrict__ g1, float* __restrict__ out) {
  extern __shared__ __align__(16) float smem[];
  float* xs  = smem;                 // [H_*VT_W] input tile
  float* g0s = xs  + H_ * VT_W;      // g0 tile, later down-scan results
  float* g1s = g0s + H_ * VT_W;      // g1 tile
  __shared__ __align__(16) float Asg[8 * VT_W];
  __shared__ __align__(16) float Bsg[8 * VT_W];

  const int    tid   = threadIdx.x;
  const int    plane = blockIdx.x / (W_ / VT_W);
  const int    w0    = (blockIdx.x % (W_ / VT_W)) * VT_W;
  const size_t pbase = (size_t)plane * PLANE + (size_t)w0;

  // Stage x, g0, g1 tile (coalesced 16B transfers; tile row = 128B).
  for (int i = tid; i < (H_ * VT_W) / 4; i += 256) {
    const int    h    = i >> 3;      // 8 float4 per 32-float row
    const int    c4   = i & 7;
    const size_t goff = pbase + (size_t)h * W_ + (size_t)c4 * 4;
    copy16_to_lds(((float4*)xs)  + i, x  + goff);
    copy16_to_lds(((float4*)g0s) + i, g0 + goff);
    copy16_to_lds(((float4*)g1s) + i, g1 + goff);
  }
  staging_fence();

  const int wave = tid >> 5;
  const int lane = tid & 31;
  const int h0   = wave * VT_SEG;

  // ---------------- down scan (h increasing, gate g0) ----------------
  {
    float A = 1.0f, Bv = 0.0f;
    for (int j = 0; j < VT_SEG; ++j) {
      const int   h = h0 + j;
      const float g = (h == 0) ? 0.0f : g0s[h * VT_W + lane];
      const float b = xs[h * VT_W + lane] * (1.0f - g);
      Bv = fmaf(g, Bv, b);
      A *= g;
    }
    Asg[wave * VT_W + lane] = A;
    Bsg[wave * VT_W + lane] = Bv;
    __syncthreads();
    if (wave == 0) {  // serial segment scan, vectorized over 32 columns
      float a = Asg[lane], b = Bsg[lane];
      for (int s = 1; s < 8; ++s) {
        const float as_ = Asg[s * VT_W + lane], bs_ = Bsg[s * VT_W + lane];
        b = fmaf(as_, b, bs_);
        a *= as_;
        Asg[s * VT_W + lane] = a;
        Bsg[s * VT_W + lane] = b;
      }
    }
    __syncthreads();
    float prev = (wave == 0) ? 0.0f : Bsg[(wave - 1) * VT_W + lane];
    for (int j = 0; j < VT_SEG; ++j) {
      const int   h = h0 + j;
      const float g = (h == 0) ? 0.0f : g0s[h * VT_W + lane];
      const float b = xs[h * VT_W + lane] * (1.0f - g);
      prev = fmaf(g, prev, b);
      g0s[h * VT_W + lane] = prev;  // stash down result over consumed gate
    }
  }
  __syncthreads();

  // ---------------- up scan (h decreasing, gate g1) ----------------
  {
    float A = 1.0f, Bv = 0.0f;
    for (int j = VT_SEG - 1; j >= 0; --j) {
      const int   h = h0 + j;
      const float g = (h == H_ - 1) ? 0.0f : g1s[h * VT_W + lane];
      const float b = xs[h * VT_W + lane] * (1.0f - g);
      Bv = fmaf(g, Bv, b);
      A *= g;
    }
    Asg[wave * VT_W + lane] = A;
    Bsg[wave * VT_W + lane] = Bv;
    __syncthreads();
    if (wave == 0) {  // reverse segment scan
      float a = Asg[7 * VT_W + lane], b = Bsg[7 * VT_W + lane];
      for (int s = 6; s >= 0; --s) {
        const float as_ = Asg[s * VT_W + lane], bs_ = Bsg[s * VT_W + lane];
        b = fmaf(as_, b, bs_);
        a *= as_;
        Asg[s * VT_W + lane] = a;
        Bsg[s * VT_W + lane] = b;
      }
    }
    __syncthreads();
    float prev = (wave == 7) ? 0.0f : Bsg[(wave + 1) * VT_W + lane];
    for (int j = VT_SEG - 1; j >= 0; --j) {
      const int   h = h0 + j;
      const float g = (h == H_ - 1) ? 0.0f : g1s[h * VT_W + lane];
      const float b = xs[h * VT_W + lane] * (1.0f - g);
      prev = fmaf(g, prev, b);
      const float d = g0s[h * VT_W + lane];
      out[pbase + (size_t)h * W_ + lane] = fmaxf(d, prev);  // single write
    }
  }
}

// ---------------------------------------------------------------------------
// Horizontal kernel: right (g2) and left (g3) scans over W; merge into out.
// Block = 8 contiguous rows; 1 wave32 per row; lane owns a 20-elem chunk.
// Wave scan of affine pairs via __shfl (wave32 => width 32).
// ---------------------------------------------------------------------------
#define HR_ROWS 8
#define HCHUNK  (W_ / 32)  // 20

__global__ __launch_bounds__(256) void nlf_horizontal(
    const float* __restrict__ x, const float* __restrict__ g2,
    const float* __restrict__ g3, float* __restrict__ out) {
  __shared__ __align__(16) float xs[HR_ROWS * W_];
  __shared__ __align__(16) float rs[HR_ROWS * W_];  // g2, then right results
  __shared__ __align__(16) float ls[HR_ROWS * W_];  // g3, then left results

  const int    tid       = threadIdx.x;
  const size_t blockBase = (size_t)blockIdx.x * HR_ROWS * W_;

  for (int i = tid; i < (HR_ROWS * W_) / 4; i += 256) {
    const size_t goff = blockBase + (size_t)i * 4;
    copy16_to_lds(((float4*)xs) + i, x  + goff);
    copy16_to_lds(((float4*)rs) + i, g2 + goff);
    copy16_to_lds(((float4*)ls) + i, g3 + goff);
  }
  // Speculative prefetch of the next block's tile -> global_prefetch_b8.
  if (blockIdx.x + 1 < gridDim.x && tid < 40) {
    const size_t nb = blockBase + (size_t)HR_ROWS * W_;
    __builtin_prefetch(x  + nb + (size_t)tid * 128, 0, 0);
    __builtin_prefetch(g2 + nb + (size_t)tid * 128, 0, 0);
    __builtin_prefetch(g3 + nb + (size_t)tid * 128, 0, 0);
  }
  staging_fence();

  const int wave = tid >> 5;
  const int lane = tid & 31;
  float* xr = xs + wave * W_;
  float* rr = rs + wave * W_;
  float* lr = ls + wave * W_;
  const int c0 = lane * HCHUNK;

  // ---------------- right scan (w increasing, gate g2) ----------------
  {
    float A = 1.0f, Bv = 0.0f;
    for (int j = 0; j < HCHUNK; ++j) {
      const int   wpos = c0 + j;
      const float g = (wpos == 0) ? 0.0f : rr[wpos];
      const float b = xr[wpos] * (1.0f - g);
      Bv = fmaf(g, Bv, b);
      A *= g;
    }
    float a = A, b = Bv;
    for (int s = 1; s < 32; s <<= 1) {  // inclusive scan, lanes ascending
      const float ap = __shfl_up(a, s, 32);
      const float bp = __shfl_up(b, s, 32);
      if (lane >= s) { b = fmaf(a, bp, b); a *= ap; }
    }
    float prev = __shfl_up(b, 1, 32);
    if (lane == 0) prev = 0.0f;
    for (int j = 0; j < HCHUNK; ++j) {
      const int   wpos = c0 + j;
      const float g  = (wpos == 0) ? 0.0f : rr[wpos];
      const float bt = xr[wpos] * (1.0f - g);
      prev = fmaf(g, prev, bt);
      rr[wpos] = prev;  // stash over consumed gate
    }
  }

  // ---------------- left scan (w decreasing, gate g3) ----------------
  {
    float A = 1.0f, Bv = 0.0f;
    for (int j = HCHUNK - 1; j >= 0; --j) {
      const int   wpos = c0 + j;
      const float g = (wpos == W_ - 1) ? 0.0f : lr[wpos];
      const float b = xr[wpos] * (1.0f - g);
      Bv = fmaf(g, Bv, b);
      A *= g;
    }
    float a = A, b = Bv;
    for (int s = 1; s < 32; s <<= 1) {  // inclusive scan, lanes descending
      const float ap = __shfl_down(a, s, 32);
      const float bp = __shfl_down(b, s, 32);
      if (lane < 32 - s) { b = fmaf(a, bp, b); a *= ap; }
    }
    float prev = __shfl_down(b, 1, 32);
    if (lane == 31) prev = 0.0f;
    for (int j = HCHUNK - 1; j >= 0; --j) {
      const int   wpos = c0 + j;
      const float g  = (wpos == W_ - 1) ? 0.0f : lr[wpos];
      const float bt = xr[wpos] * (1.0f - g);
      prev = fmaf(g, prev, bt);
      lr[wpos] = prev;  // stash over consumed gate
    }
  }
  __syncthreads();

  // Final fully-coalesced merge with the vertical result already in out.
  for (int i = tid; i < HR_ROWS * W_; i += 256) {
    const float v = out[blockBase + i];
    out[blockBase + i] = fmaxf(v, fmaxf(rs[i], ls[i]));
  }
}

// ---------------------------------------------------------------------------
extern "C" void kernel_launch(void* const* d_in, const int* in_sizes, int n_in,
                              void* d_out, int out_size, void* d_ws,
                              size_t ws_size, hipStream_t stream) {
  (void)in_sizes; (void)n_in; (void)out_size; (void)d_ws; (void)ws_size;
  const float* x  = (const float*)d_in[0];
  const float* g0 = (const float*)d_in[1];
  const float* g1 = (const float*)d_in[2];
  const float* g2 = (const float*)d_in[3];
  const float* g3 = (const float*)d_in[4];
  float*       out = (float*)d_out;

  // 3 * 320 * 32 * 4B = 120 KB dynamic LDS (CDNA5 WGP has 320 KB).
  const size_t vshmem = (size_t)3 * H_ * VT_W * sizeof(float);
  (void)hipFuncSetAttribute((const void*)nlf_vertical,
                            hipFuncAttributeMaxDynamicSharedMemorySize,
                            (int)vshmem);

  nlf_vertical<<<dim3(NPLANES * (W_ / VT_W)), dim3(256), vshmem, stream>>>(
      x, g0, g1, out);
  nlf_horizontal<<<dim3(NROWS / HR_ROWS), dim3(256), 0, stream>>>(
      x, g2, g3, out);
}